// InitNodeSelectionModel_25872882991239
// MI455X (gfx1250) — compile-verified
//
#include <hip/hip_runtime.h>
#include <hip/hip_bf16.h>
#include <math.h>

// ---------------- problem constants (match reference) ----------------
static constexpr int B   = 8;
static constexpr int N   = 512;
static constexpr int E   = 8192;
static constexpr int G   = 1024;
static constexpr int HD  = 1024;
static constexpr int FIN = 20;
static constexpr int K2  = 2 * HD;   // 2048 (fc_w inner dim)
static constexpr int NO  = 128;      // fc_w output features
static constexpr float BN_EPS = 1e-5f;

typedef float v2f __attribute__((ext_vector_type(2)));
typedef float v8f __attribute__((ext_vector_type(8)));

// ---------------------------------------------------------------------
// K1: per-graph in-degree (LDS int atomics)
// grid = B blocks, 256 threads
__global__ void deg_kernel(const int* __restrict__ dst, int* __restrict__ deg) {
    __shared__ int sdeg[N];
    const int b = blockIdx.x;
    for (int i = threadIdx.x; i < N; i += blockDim.x) sdeg[i] = 0;
    __syncthreads();
    for (int e = threadIdx.x; e < E; e += blockDim.x)
        atomicAdd(&sdeg[dst[b * E + e]], 1);
    __syncthreads();
    for (int i = threadIdx.x; i < N; i += blockDim.x) deg[b * N + i] = sdeg[i];
}

// ---------------------------------------------------------------------
// K2: X0[b][n][f] = h[b][n][:] . fc_init_w[f][:] + fc_init_b[f]
// grid = B*N*HD/256
__global__ void fc_init_kernel(const float* __restrict__ h,
                               const float* __restrict__ w,
                               const float* __restrict__ bias,
                               float* __restrict__ x0) {
    const int idx = blockIdx.x * blockDim.x + threadIdx.x;   // [0, B*N*HD)
    const int f  = idx & (HD - 1);
    const int bn = idx >> 10;                                // b*N + n
    const float* __restrict__ hrow = h + (size_t)bn * FIN;
    const float* __restrict__ wrow = w + (size_t)f * FIN;
    float acc = bias[f];
    #pragma unroll
    for (int t = 0; t < FIN; ++t) acc = fmaf(hrow[t], wrow[t], acc);
    x0[idx] = acc;
}

// ---------------------------------------------------------------------
// K3: segment-sum over edges: sums[b][dst][f] = sum_e x[b][src[e]][f]
// One block per (feature-chunk of 32, graph). 64KB LDS tile [N][32].
// Lane index == feature -> bank-conflict-free ds_add_f32.
// grid = dim3(HD/32, B), 256 threads (8 waves)
__global__ void segsum_kernel(const int* __restrict__ src,
                              const int* __restrict__ dst,
                              const float* __restrict__ x,
                              float* __restrict__ sums) {
    __shared__ float ssum[N * 32];                           // 64 KB
    const int chunk = blockIdx.x;
    const int b     = blockIdx.y;
    const int lane  = threadIdx.x & 31;
    const int wid   = threadIdx.x >> 5;

    for (int i = threadIdx.x; i < N * 32; i += blockDim.x) ssum[i] = 0.f;
    __syncthreads();

    const int fbase = chunk * 32;
    for (int e = wid; e < E; e += 8) {
        const int sn = src[b * E + e];
        const int dn = dst[b * E + e];
        const float v = x[((size_t)(b * N + sn)) * HD + fbase + lane];
        atomicAdd(&ssum[dn * 32 + lane], v);
    }
    __syncthreads();

    for (int i = threadIdx.x; i < N * 32; i += blockDim.x) {
        const int n = i >> 5;
        const int l = i & 31;
        sums[((size_t)(b * N + n)) * HD + fbase + l] = ssum[i];
    }
}

// ---------------------------------------------------------------------
// K4/K6: t = x + (deg>0 ? sums/deg : 0); batch-norm over nodes; ReLU.
// Optionally emit qemb (node mean of output).
// grid = dim3(HD/256, B), 256 threads; thread == feature -> coalesced
__global__ void bn_gin_kernel(const float* __restrict__ x,
                              const float* __restrict__ sums,
                              const int* __restrict__ deg,
                              const float* __restrict__ gamma,
                              const float* __restrict__ beta,
                              float* __restrict__ out,
                              float* __restrict__ qemb) {
    const int b = blockIdx.y;
    const int f = blockIdx.x * blockDim.x + threadIdx.x;

    float sum = 0.f, sq = 0.f;
    for (int n = 0; n < N; ++n) {
        const size_t idx = ((size_t)(b * N + n)) * HD + f;
        const int dg = deg[b * N + n];
        float t = x[idx];
        if (dg > 0) t += sums[idx] / (float)dg;
        sum += t;
        sq  = fmaf(t, t, sq);
    }
    const float mean = sum * (1.0f / N);
    const float var  = sq * (1.0f / N) - mean * mean;   // biased, matches ref
    const float scale = rsqrtf(var + BN_EPS) * gamma[f];
    const float shift = beta[f] - mean * scale;

    float qacc = 0.f;
    for (int n = 0; n < N; ++n) {
        const size_t idx = ((size_t)(b * N + n)) * HD + f;
        const int dg = deg[b * N + n];
        float t = x[idx];
        if (dg > 0) t += sums[idx] / (float)dg;
        const float y = fmaxf(fmaf(t, scale, shift), 0.f);
        out[idx] = y;
        qacc += y;
    }
    if (qemb) qemb[b * HD + f] = qacc * (1.0f / N);
}

// ---------------------------------------------------------------------
// K5a: qproj[b][j] = qemb[b][:] . fc_w[j][:HD]
// The k<HD half of the big GEMM is row-invariant (qemb broadcast), so
// reduce it once per graph instead of 1024x. One wave per (b,j).
// grid = B*NO/8 = 128 blocks, 256 threads
__global__ void qproj_kernel(const float* __restrict__ qemb,
                             const float* __restrict__ fcw,
                             float* __restrict__ qproj) {
    const int lane = threadIdx.x & 31;
    const int wid  = threadIdx.x >> 5;
    const int bj   = blockIdx.x * 8 + wid;          // [0, B*NO)
    const int b    = bj >> 7;                       // / NO
    const int j    = bj & (NO - 1);

    const float* __restrict__ q = qemb + (size_t)b * HD;
    const float* __restrict__ w = fcw + (size_t)j * K2;
    float acc = 0.f;
    for (int k = lane; k < HD; k += 32) acc = fmaf(q[k], w[k], acc);
    #pragma unroll
    for (int off = 16; off; off >>= 1) acc += __shfl_xor(acc, off);
    if (lane == 0) qproj[bj] = acc;
}

// ---------------------------------------------------------------------
// K5b: Z[b][g][j] = qproj[b][j] + fc_b[j]
//                 + sum_{k<HD} gEmb[g][k] * fc_w[j][HD+k]
//     gEmb[g] = allDBGEmb[gPos[b][g]]
// fp32 WMMA 16x16x4 over the gathered half only (K=1024).
// Each wave: 16(M) x 64(N) tile, 4 accumulators.
// grid = B*(G/16)*(NO/64)/8 = 128 blocks, 256 threads (8 waves)
__global__ void gemm_kernel(const float* __restrict__ qproj,
                            const int* __restrict__ gPos,
                            const float* __restrict__ emb,
                            const float* __restrict__ fcw,
                            const float* __restrict__ fcb,
                            float* __restrict__ Z) {
    const int lane = threadIdx.x & 31;
    const int wid  = threadIdx.x >> 5;
    const int gw   = blockIdx.x * 8 + wid;         // [0, 1024)
    const int b    = gw >> 7;                      // / (64*2)
    const int rem  = gw & 127;
    const int g0   = (rem >> 1) * 16;              // M tile base
    const int j0   = (rem & 1) * 64;               // N half base

    const int lo = lane & 15;
    const int hi = lane >> 4;

    // A-matrix per-lane row: M = lane&15 (ISA 16x4 f32 layout)
    const int m_a = g0 + lo;
    const float* __restrict__ grow = emb + (size_t)gPos[b * G + m_a] * HD;

    // Warm the HBM gather path for this lane's 4KB embedding row while
    // we set up accumulators (global_prefetch_b8).
    __builtin_prefetch(grow, 0, 0);
    __builtin_prefetch(grow + 256, 0, 0);
    __builtin_prefetch(grow + 512, 0, 0);
    __builtin_prefetch(grow + 768, 0, 0);

    // B/C/D per-lane column: N = lane&15; fold bias + qemb projection.
    v8f acc[4];
    #pragma unroll
    for (int t = 0; t < 4; ++t) {
        const int n = j0 + 16 * t + lo;
        const float c0 = fcb[n] + qproj[b * NO + n];
        #pragma unroll
        for (int r = 0; r < 8; ++r) acc[t][r] = c0;
    }

    const float* __restrict__ wbase = fcw + HD;     // second K-half of fc_w
    #pragma unroll 2
    for (int k0 = 0; k0 < HD; k0 += 4) {
        const int koff = k0 + 2 * hi;               // even -> 8B aligned
        const float2 av = *(const float2*)(grow + koff);
        v2f a; a[0] = av.x; a[1] = av.y;
        #pragma unroll
        for (int t = 0; t < 4; ++t) {
            const int n_b = j0 + 16 * t + lo;
            const float2 bv = *(const float2*)(wbase + (size_t)n_b * K2 + koff);
            v2f bb; bb[0] = bv.x; bb[1] = bv.y;
            acc[t] = __builtin_amdgcn_wmma_f32_16x16x4_f32(
                false, a, false, bb, (short)0, acc[t], false, false);
        }
    }

    #pragma unroll
    for (int t = 0; t < 4; ++t) {
        #pragma unroll
        for (int r = 0; r < 8; ++r) {
            const int m = g0 + r + 8 * hi;
            const int n = j0 + 16 * t + lo;
            Z[((size_t)(b * G + m)) * NO + n] = acc[t][r];
        }
    }
}

// ---------------------------------------------------------------------
// K7: per-graph BN3 batch stats over G rows
// grid = B blocks, 128 threads (thread == output feature j)
__global__ void bn3stats_kernel(const float* __restrict__ Z,
                                float* __restrict__ mu,
                                float* __restrict__ var) {
    const int b = blockIdx.x;
    const int j = threadIdx.x;
    float sum = 0.f, sq = 0.f;
    for (int g = 0; g < G; ++g) {
        const float z = Z[((size_t)(b * G + g)) * NO + j];
        sum += z;
        sq  = fmaf(z, z, sq);
    }
    const float m = sum * (1.0f / G);
    mu[b * NO + j]  = m;
    var[b * NO + j] = sq * (1.0f / G) - m * m;
}

// ---------------------------------------------------------------------
// K8: out[b][g] = sigmoid( relu(bn3(Z[b][g][:])) . fc2_w + fc2_b )
// one wave (32 lanes) per row; grid = B*G/8 blocks, 256 threads
__global__ void final_kernel(const float* __restrict__ Z,
                             const float* __restrict__ mu,
                             const float* __restrict__ var,
                             const float* __restrict__ g3,
                             const float* __restrict__ b3,
                             const float* __restrict__ w2,
                             const float* __restrict__ b2,
                             float* __restrict__ out) {
    const int lane = threadIdx.x & 31;
    const int wid  = threadIdx.x >> 5;
    const int row  = blockIdx.x * 8 + wid;          // [0, B*G)
    const int b    = row >> 10;

    float acc = 0.f;
    #pragma unroll
    for (int jj = 0; jj < 4; ++jj) {
        const int j = lane + (jj << 5);
        const float z = Z[(size_t)row * NO + j];
        const float zn = (z - mu[b * NO + j]) * rsqrtf(var[b * NO + j] + BN_EPS)
                         * g3[j] + b3[j];
        acc = fmaf(fmaxf(zn, 0.f), w2[j], acc);
    }
    #pragma unroll
    for (int off = 16; off; off >>= 1) acc += __shfl_xor(acc, off);
    if (lane == 0) out[row] = 1.0f / (1.0f + __expf(-(acc + b2[0])));
}

// ---------------------------------------------------------------------
extern "C" void kernel_launch(void* const* d_in, const int* in_sizes, int n_in,
                              void* d_out, int out_size, void* d_ws, size_t ws_size,
                              hipStream_t stream) {
    const float* h      = (const float*)d_in[0];
    const int*   src    = (const int*)  d_in[1];
    const int*   dst    = (const int*)  d_in[2];
    const int*   gPos   = (const int*)  d_in[3];
    const float* emb    = (const float*)d_in[4];
    const float* w0     = (const float*)d_in[5];
    const float* b0     = (const float*)d_in[6];
    const float* bn1g   = (const float*)d_in[7];
    const float* bn1b   = (const float*)d_in[8];
    const float* bn2g   = (const float*)d_in[9];
    const float* bn2b   = (const float*)d_in[10];
    const float* fcw    = (const float*)d_in[11];
    const float* fcb    = (const float*)d_in[12];
    const float* bn3g   = (const float*)d_in[13];
    const float* bn3b   = (const float*)d_in[14];
    const float* fc2w   = (const float*)d_in[15];
    const float* fc2b   = (const float*)d_in[16];
    float* out = (float*)d_out;

    // workspace carve-up (floats)
    const size_t XN = (size_t)B * N * HD;           // 4,194,304
    float* X0    = (float*)d_ws;
    float* X1    = X0 + XN;
    float* SUMS  = X1 + XN;
    float* Zbuf  = SUMS + XN;                       // B*G*NO = 1,048,576
    float* QEMB  = Zbuf + (size_t)B * G * NO;       // B*HD
    float* QPROJ = QEMB + (size_t)B * HD;           // B*NO
    float* MU3   = QPROJ + (size_t)B * NO;          // B*NO
    float* VAR3  = MU3 + (size_t)B * NO;            // B*NO
    int*   DEG   = (int*)(VAR3 + (size_t)B * NO);   // B*N

    deg_kernel<<<B, 256, 0, stream>>>(dst, DEG);
    fc_init_kernel<<<(int)(XN / 256), 256, 0, stream>>>(h, w0, b0, X0);

    segsum_kernel<<<dim3(HD / 32, B), 256, 0, stream>>>(src, dst, X0, SUMS);
    bn_gin_kernel<<<dim3(HD / 256, B), 256, 0, stream>>>(X0, SUMS, DEG, bn1g, bn1b, X1, nullptr);

    segsum_kernel<<<dim3(HD / 32, B), 256, 0, stream>>>(src, dst, X1, SUMS);
    bn_gin_kernel<<<dim3(HD / 256, B), 256, 0, stream>>>(X1, SUMS, DEG, bn2g, bn2b, X0, QEMB);

    qproj_kernel<<<B * NO / 8, 256, 0, stream>>>(QEMB, fcw, QPROJ);
    gemm_kernel<<<B * (G / 16) * (NO / 64) / 8, 256, 0, stream>>>(QPROJ, gPos, emb, fcw, fcb, Zbuf);

    bn3stats_kernel<<<B, NO, 0, stream>>>(Zbuf, MU3, VAR3);
    final_kernel<<<B * G / 8, 256, 0, stream>>>(Zbuf, MU3, VAR3, bn3g, bn3b, fc2w, fc2b, out);
}